// GConvLayer_11312943858313
// MI455X (gfx1250) — compile-verified
//
#include <hip/hip_runtime.h>
#include <hip/hip_bf16.h>

typedef float v2f __attribute__((ext_vector_type(2)));
typedef float v8f __attribute__((ext_vector_type(8)));

// ---------------- degree / normalization ----------------

__global__ void k_deg_init(float* __restrict__ deg, int n) {
  int i = blockIdx.x * blockDim.x + threadIdx.x;
  if (i < n) deg[i] = 1.0f;  // self-loop contributes 1
}

__global__ void k_deg_count(const int* __restrict__ col, float* __restrict__ deg, int e) {
  int i = blockIdx.x * blockDim.x + threadIdx.x;
  if (i < e) atomicAdd(&deg[col[i]], 1.0f);
}

__global__ void k_deg_rsqrt(const float* __restrict__ deg, float* __restrict__ dinv, int n) {
  int i = blockIdx.x * blockDim.x + threadIdx.x;
  if (i < n) dinv[i] = rsqrtf(deg[i]);  // deg >= 1 always (self-loops)
}

// ---------------- dense GEMM: xw = x @ W^T via V_WMMA_F32_16X16X4_F32 ----------------
// One wave computes a 16-node x 128-feature strip (8 WMMA tiles, A reused).
// Software-pipelined: fragments for k+4 are loaded into fresh registers while
// the 8 WMMAs for k execute, so waits are partial loadcnt thresholds instead
// of full drains.
// A (16x4 f32): lane L holds row m=L%16; VGPR v holds K = (L>>4)*2 + v  -> float2 load
// B (4x16 f32): lane L holds col n=L%16; VGPR v holds K = (L>>4)*2 + v  -> float2 of W[n][k..k+1]
// C (16x16 f32): VGPR v: lanes 0-15 -> M=v, lanes 16-31 -> M=v+8; N = L%16

__global__ void k_gemm_wmma(const float* __restrict__ x,
                            const float* __restrict__ W,
                            float* __restrict__ xw, int ntiles) {
  const int lane   = threadIdx.x & 31;
  const int waveIb = threadIdx.x >> 5;
  const int tile   = blockIdx.x * (blockDim.x >> 5) + waveIb;
  if (tile >= ntiles) return;  // whole-wave uniform: EXEC stays all-ones for WMMA

  const int m     = lane & 15;
  const int khalf = lane >> 4;  // 0 or 1

  const float* xrow  = x + (size_t)(tile * 16 + m) * 128 + khalf * 2;
  const float* wbase = W + (size_t)m * 128 + khalf * 2;  // tile t adds t*16*128 = t*2048

  v8f acc[8] = {};

  // prologue: fragments for k0 = 0
  v2f a_cur = *(const v2f*)(xrow);
  v2f b_cur[8];
#pragma unroll
  for (int t = 0; t < 8; ++t) b_cur[t] = *(const v2f*)(wbase + t * 2048);

#pragma unroll
  for (int k0 = 0; k0 < 128; k0 += 4) {
    v2f a_nxt = a_cur;
    v2f b_nxt[8];
#pragma unroll
    for (int t = 0; t < 8; ++t) b_nxt[t] = b_cur[t];
    if (k0 + 4 < 128) {  // prefetch next k-step while WMMAs below execute
      a_nxt = *(const v2f*)(xrow + k0 + 4);
#pragma unroll
      for (int t = 0; t < 8; ++t)
        b_nxt[t] = *(const v2f*)(wbase + t * 2048 + k0 + 4);
    }
#pragma unroll
    for (int t = 0; t < 8; ++t) {
      acc[t] = __builtin_amdgcn_wmma_f32_16x16x4_f32(
          /*neg_a=*/false, a_cur, /*neg_b=*/false, b_cur[t],
          /*c_mod=*/(short)0, acc[t], /*reuse_a=*/false, /*reuse_b=*/false);
    }
    a_cur = a_nxt;
#pragma unroll
    for (int t = 0; t < 8; ++t) b_cur[t] = b_nxt[t];
  }

  // store: element (v, lane): row = tile*16 + khalf*8 + v, col = t*16 + m
  float* outbase = xw + (size_t)(tile * 16 + khalf * 8) * 128 + m;
#pragma unroll
  for (int t = 0; t < 8; ++t) {
#pragma unroll
    for (int v = 0; v < 8; ++v) {
      outbase[(size_t)v * 128 + t * 16] = acc[t][v];
    }
  }
}

// scalar tail for node counts not divisible by 16 (unused for N=100000)
__global__ void k_gemm_tail(const float* __restrict__ x, const float* __restrict__ W,
                            float* __restrict__ xw, int n, int start) {
  int i = blockIdx.x * blockDim.x + threadIdx.x;
  int total = (n - start) * 128;
  if (i >= total) return;
  int node = start + i / 128;
  int o = i & 127;
  const float* xr = x + (size_t)node * 128;
  const float* wr = W + (size_t)o * 128;
  float s = 0.f;
  for (int k = 0; k < 128; ++k) s += xr[k] * wr[k];
  xw[(size_t)node * 128 + o] = s;
}

// ---------------- out = xw * dinv^2 + b   (self-loop term, full init of d_out) ----------------

__global__ void k_init_out(const float* __restrict__ xw, const float* __restrict__ dinv,
                           const float* __restrict__ bias, float* __restrict__ out, int n) {
  int i = blockIdx.x * blockDim.x + threadIdx.x;  // one float4 per thread
  int total = n * 32;                             // 128/4 chunks per node
  if (i >= total) return;
  int node = i >> 5;
  int f = (i & 31) * 4;
  float d = dinv[node];
  float s = d * d;
  const float4 v = *(const float4*)(xw + (size_t)node * 128 + f);
  const float4 bb = *(const float4*)(bias + f);
  float4 o;
  o.x = v.x * s + bb.x;
  o.y = v.y * s + bb.y;
  o.z = v.z * s + bb.z;
  o.w = v.w * s + bb.w;
  *(float4*)(out + (size_t)node * 128 + f) = o;
}

// ---------------- wave-per-edge gather / atomic scatter ----------------

__global__ void k_scatter(const float* __restrict__ xw,
                          const int* __restrict__ row, const int* __restrict__ col,
                          const float* __restrict__ dinv,
                          float* __restrict__ out, int e) {
  const int lane = threadIdx.x & 31;
  const int edge = (blockIdx.x * blockDim.x + threadIdx.x) >> 5;
  if (edge >= e) return;
  const int r = row[edge];
  const int c = col[edge];
  const float nrm = dinv[r] * dinv[c];
  const float4 v = *(const float4*)(xw + (size_t)r * 128 + lane * 4);
  float* dst = out + (size_t)c * 128 + lane * 4;
  atomicAdd(dst + 0, v.x * nrm);
  atomicAdd(dst + 1, v.y * nrm);
  atomicAdd(dst + 2, v.z * nrm);
  atomicAdd(dst + 3, v.w * nrm);
}

// ---------------- launcher ----------------

extern "C" void kernel_launch(void* const* d_in, const int* in_sizes, int n_in,
                              void* d_out, int out_size, void* d_ws, size_t ws_size,
                              hipStream_t stream) {
  const float* x  = (const float*)d_in[0];
  const int* ei   = (const int*)d_in[1];
  const float* W  = (const float*)d_in[2];
  const float* b  = (const float*)d_in[3];
  float* out      = (float*)d_out;

  const int N = in_sizes[0] / 128;
  const int E = in_sizes[1] / 2;
  const int* row = ei;       // source nodes
  const int* col = ei + E;   // target nodes

  // workspace layout: xw [N*128] | deg [N] | dinv [N]
  float* xw   = (float*)d_ws;
  float* deg  = xw + (size_t)N * 128;
  float* dinv = deg + N;

  // 1-3: degree + D^{-1/2}
  k_deg_init<<<(N + 255) / 256, 256, 0, stream>>>(deg, N);
  k_deg_count<<<(E + 255) / 256, 256, 0, stream>>>(col, deg, E);
  k_deg_rsqrt<<<(N + 255) / 256, 256, 0, stream>>>(deg, dinv, N);

  // 4: xw = x @ W^T (WMMA f32)
  const int ntiles = N / 16;
  if (ntiles > 0)
    k_gemm_wmma<<<(ntiles + 7) / 8, 256, 0, stream>>>(x, W, xw, ntiles);
  const int rem = N % 16;
  if (rem > 0)
    k_gemm_tail<<<(rem * 128 + 255) / 256, 256, 0, stream>>>(x, W, xw, N, ntiles * 16);

  // 5: self-loop term + bias (fully initializes out)
  k_init_out<<<(N * 32 + 255) / 256, 256, 0, stream>>>(xw, dinv, b, out, N);

  // 6: edge scatter with f32 atomics
  k_scatter<<<(E + 7) / 8, 256, 0, stream>>>(xw, row, col, dinv, out, E);
}